// SimpleSAGE_9208409883074
// MI455X (gfx1250) — compile-verified
//
#include <hip/hip_runtime.h>
#include <hip/hip_bf16.h>

typedef __attribute__((ext_vector_type(2))) float v2f;
typedef __attribute__((ext_vector_type(8))) float v8f;

#define NOUT 256
#define LN_EPS 1e-5f

// ---------------------------------------------------------------------------
// Utility kernels
// ---------------------------------------------------------------------------
__global__ __launch_bounds__(256) void zero_kernel(float* __restrict__ p, int n) {
    int i = blockIdx.x * 256 + threadIdx.x;
    if (i < n) p[i] = 0.0f;
}

__global__ __launch_bounds__(256) void degree_kernel(const int* __restrict__ dst,
                                                     float* __restrict__ deg, int E) {
    int e = blockIdx.x * 256 + threadIdx.x;
    if (e < E) atomicAdd(&deg[dst[e]], 1.0f);
}

__global__ __launch_bounds__(256) void recip_kernel(float* __restrict__ deg, int n) {
    int i = blockIdx.x * 256 + threadIdx.x;
    if (i < n) deg[i] = 1.0f / deg[i];   // self loops guarantee deg >= 1
}

// Scatter-add neighbor features: agg[dst] += h[src].  One (edge, feature) per lane.
// h / agg are L2-resident (51 MB << 192 MB L2) so the float atomics resolve in L2.
template <int LOGF>
__global__ __launch_bounds__(256) void scatter_add_kernel(const float* __restrict__ h,
                                                          const int* __restrict__ src,
                                                          const int* __restrict__ dst,
                                                          float* __restrict__ agg, int E) {
    constexpr int F = 1 << LOGF;
    int gid = blockIdx.x * 256 + threadIdx.x;
    int e = gid >> LOGF;
    if (e >= E) return;
    int f = gid & (F - 1);
    atomicAdd(&agg[(size_t)dst[e] * F + f], h[(size_t)src[e] * F + f]);
}

// ---------------------------------------------------------------------------
// Fused SAGE layer:  out = ELU(LayerNorm(h@Ws + (agg*deginv)@Wn + b))
// One block = 32 output rows x 256 output cols.  8 waves; wave w owns columns
// [32w, 32w+32) for BOTH 16-row subtiles => 4 fp32 WMMA accumulators per wave,
// so each B fetch feeds 4 WMMAs (2x less weight traffic than a 16-row tile).
// H tile is staged to LDS with the CDNA5 async-to-LDS path (ASYNCcnt);
// AGG tile goes through VGPRs because it is scaled by deginv on the way in.
// ---------------------------------------------------------------------------
template <int K>
__global__ __launch_bounds__(256, 2)
void sage_layer_kernel(const float* __restrict__ H,      // nrows x K
                       const float* __restrict__ AGG,    // nrows x K (unscaled sums)
                       const float* __restrict__ deginv, // nrows
                       const float* __restrict__ Ws,     // K x 256
                       const float* __restrict__ Wn,     // K x 256
                       const float* __restrict__ bias,   // 256
                       const float* __restrict__ gamma,  // 256
                       const float* __restrict__ beta,   // 256
                       float* __restrict__ Out,          // nrows x 256
                       int nrows)
{
    constexpr int KP  = K + 4;             // padded LDS row stride (even -> b64 aligned)
    constexpr int CPR = K / 4;             // 16-byte chunks per row
    __shared__ float smem[2 * 32 * KP];    // A tiles; later reused as 32x264 out tile
    __shared__ float red[32][8][2];        // LN partial sums / sumsq
    __shared__ float mv[32][2];            // per-row mean, rstd

    float* ldsH = smem;
    float* ldsA = smem + 32 * KP;
    const int tid  = threadIdx.x;
    const int row0 = blockIdx.x * 32;

    // ---- async stage of the 32xK H tile into LDS (ASYNCcnt path) ----
    for (int idx = tid; idx < 32 * CPR; idx += 256) {
        const int r = idx / CPR;
        const int c = (idx - r * CPR) * 4;
        int row = row0 + r; if (row >= nrows) row = nrows - 1;   // clamp tail
        const unsigned long long ga =
            (unsigned long long)(uintptr_t)(H + (size_t)row * K + c);
        const unsigned lo = (unsigned)(uintptr_t)(const void*)(ldsH + r * KP + c);
        asm volatile("global_load_async_to_lds_b128 %0, %1, off"
                     :: "v"(lo), "v"(ga) : "memory");
    }
    // ---- regular stage of the scaled AGG tile ----
    for (int idx = tid; idx < 32 * CPR; idx += 256) {
        const int r = idx / CPR;
        const int c = (idx - r * CPR) * 4;
        int row = row0 + r; if (row >= nrows) row = nrows - 1;
        const float4 u = *(const float4*)(AGG + (size_t)row * K + c);
        const float sc = deginv[row];
        float* q = ldsA + r * KP + c;
        q[0] = u.x * sc; q[1] = u.y * sc; q[2] = u.z * sc; q[3] = u.w * sc;
    }
    asm volatile("s_wait_asynccnt 0x0" ::: "memory");
    __syncthreads();

    const int lane = tid & 31;
    const int w    = tid >> 5;       // wave id 0..7
    const int half = lane >> 4;      // K-pair select (ISA A/B lane layout)
    const int lr   = lane & 15;      // A: row within tile / B,D: col within tile
    const int n0   = w * 32 + lr;

    v8f acc00 = {}, acc01 = {}, acc10 = {}, acc11 = {};

    #pragma unroll 1
    for (int pass = 0; pass < 2; ++pass) {
        const float* As = pass ? ldsA : ldsH;
        const float* W  = pass ? Wn : Ws;
        #pragma unroll 4
        for (int kk = 0; kk < K; kk += 4) {
            const int kb = kk + 2 * half;
            // A 16x4: lane lr holds row lr, K elements kb, kb+1 (two M subtiles)
            v2f a0 = *(const v2f*)(As + lr * KP + kb);
            v2f a1 = *(const v2f*)(As + (16 + lr) * KP + kb);
            // B 4x16: lane lr holds col n0, K rows kb, kb+1 (two N subtiles)
            v2f b0, b1;
            b0.x = W[kb * NOUT + n0];        b0.y = W[(kb + 1) * NOUT + n0];
            b1.x = W[kb * NOUT + n0 + 16];   b1.y = W[(kb + 1) * NOUT + n0 + 16];
            acc00 = __builtin_amdgcn_wmma_f32_16x16x4_f32(false, a0, false, b0,
                                                          (short)0, acc00, false, false);
            acc01 = __builtin_amdgcn_wmma_f32_16x16x4_f32(false, a0, false, b1,
                                                          (short)0, acc01, false, false);
            acc10 = __builtin_amdgcn_wmma_f32_16x16x4_f32(false, a1, false, b0,
                                                          (short)0, acc10, false, false);
            acc11 = __builtin_amdgcn_wmma_f32_16x16x4_f32(false, a1, false, b1,
                                                          (short)0, acc11, false, false);
        }
    }
    __syncthreads();   // done reading A tiles; smem is reused as the output tile

    // D layout: VGPR i -> row i + 8*half, col lr (within each 16x16 tile)
    float* outT = smem;                    // 32 x 264 (padded)
    #pragma unroll
    for (int i = 0; i < 8; ++i) {
        const int r  = i + 8 * half;
        const float bl = bias[w * 32 + lr];
        const float bh = bias[w * 32 + 16 + lr];
        outT[r * 264 + w * 32 + lr]             = acc00[i] + bl;
        outT[r * 264 + w * 32 + 16 + lr]        = acc01[i] + bh;
        outT[(16 + r) * 264 + w * 32 + lr]      = acc10[i] + bl;
        outT[(16 + r) * 264 + w * 32 + 16 + lr] = acc11[i] + bh;
    }
    __syncthreads();

    // ---- LayerNorm reduction: 8 threads per row, 32 elements each ----
    const int r = tid >> 3;          // 0..31
    const int j = tid & 7;           // 0..7
    float s = 0.0f, s2 = 0.0f;
    #pragma unroll
    for (int i = 0; i < 32; ++i) {
        const float x = outT[r * 264 + j * 32 + i];
        s += x; s2 += x * x;
    }
    red[r][j][0] = s;
    red[r][j][1] = s2;
    __syncthreads();

    if (tid < 32) {
        float m = 0.0f, q = 0.0f;
        #pragma unroll
        for (int j2 = 0; j2 < 8; ++j2) { m += red[tid][j2][0]; q += red[tid][j2][1]; }
        m *= (1.0f / NOUT);
        q = q * (1.0f / NOUT) - m * m;         // E[x^2] - mean^2
        mv[tid][0] = m;
        mv[tid][1] = rsqrtf(q + LN_EPS);
    }
    __syncthreads();

    // ---- normalize + affine + ELU + store (guard tail rows) ----
    if (row0 + r < nrows) {
        const float m  = mv[r][0];
        const float rs = mv[r][1];
        const size_t rowg = (size_t)(row0 + r) * NOUT;
        #pragma unroll
        for (int i = 0; i < 32; ++i) {
            const int col = j * 32 + i;
            float x = (outT[r * 264 + col] - m) * rs * gamma[col] + beta[col];
            x = (x > 0.0f) ? x : expm1f(x);    // jax.nn.elu, alpha = 1
            Out[rowg + col] = x;
        }
    }
}

// ---------------------------------------------------------------------------
// Per-graph mean readout: block g, thread f, loop rows of graph g.
// ---------------------------------------------------------------------------
__global__ __launch_bounds__(256) void readout_kernel(const float* __restrict__ h,
                                                      float* __restrict__ hg, int npg) {
    const int g = blockIdx.x;
    const int f = threadIdx.x;
    const float* base = h + (size_t)g * npg * NOUT + f;
    float s = 0.0f;
    for (int r = 0; r < npg; ++r) s += base[(size_t)r * NOUT];
    hg[g * NOUT + f] = s * (1.0f / (float)npg);
}

// ---------------------------------------------------------------------------
extern "C" void kernel_launch(void* const* d_in, const int* in_sizes, int n_in,
                              void* d_out, int out_size, void* d_ws, size_t ws_size,
                              hipStream_t stream) {
    const float* features = (const float*)d_in[0];
    const int*   esrc     = (const int*)d_in[1];
    const int*   edst     = (const int*)d_in[2];
    const int    E        = in_sizes[1];
    const int    N        = in_sizes[3];          // graph_ids length == node count
    const int    G        = 16;
    const int    npg      = N / G;

    const float* Ws0 = (const float*)d_in[4];
    const float* Wn0 = (const float*)d_in[5];
    const float* b0  = (const float*)d_in[6];
    const float* g0  = (const float*)d_in[7];
    const float* be0 = (const float*)d_in[8];
    const float* Ws1 = (const float*)d_in[9];
    const float* Wn1 = (const float*)d_in[10];
    const float* b1  = (const float*)d_in[11];
    const float* g1  = (const float*)d_in[12];
    const float* be1 = (const float*)d_in[13];
    const float* Ws2 = (const float*)d_in[14];
    const float* Wn2 = (const float*)d_in[15];
    const float* b2  = (const float*)d_in[16];
    const float* g2  = (const float*)d_in[17];
    const float* be2 = (const float*)d_in[18];

    // workspace layout (floats): deginv | agg (N*256) | h2 (N*256)
    float* ws     = (float*)d_ws;
    float* deginv = ws;
    float* agg    = ws + ((N + 255) & ~255);
    float* h2     = agg + (size_t)N * NOUT;

    // h1 and h3 live in d_out's first N*256 region; hg follows.
    float* h1 = (float*)d_out;
    float* h3 = (float*)d_out;
    float* hg = (float*)d_out + (size_t)N * NOUT;

    const int nfeat   = N * NOUT;
    const int mblocks = (N + 31) / 32;

    // 1) degree -> inverse degree
    zero_kernel<<<(N + 255) / 256, 256, 0, stream>>>(deginv, N);
    degree_kernel<<<(E + 255) / 256, 256, 0, stream>>>(edst, deginv, E);
    recip_kernel<<<(N + 255) / 256, 256, 0, stream>>>(deginv, N);

    // 2) layer 0  (K = 128): features -> h1
    zero_kernel<<<(N * 128 + 255) / 256, 256, 0, stream>>>(agg, N * 128);
    scatter_add_kernel<7><<<((E << 7) + 255) / 256, 256, 0, stream>>>(features, esrc, edst, agg, E);
    sage_layer_kernel<128><<<mblocks, 256, 0, stream>>>(features, agg, deginv,
                                                        Ws0, Wn0, b0, g0, be0, h1, N);

    // 3) layer 1  (K = 256): h1 -> h2
    zero_kernel<<<(nfeat + 255) / 256, 256, 0, stream>>>(agg, nfeat);
    scatter_add_kernel<8><<<((E << 8) + 255) / 256, 256, 0, stream>>>(h1, esrc, edst, agg, E);
    sage_layer_kernel<256><<<mblocks, 256, 0, stream>>>(h1, agg, deginv,
                                                        Ws1, Wn1, b1, g1, be1, h2, N);

    // 4) layer 2  (K = 256): h2 -> h3 (= d_out)
    zero_kernel<<<(nfeat + 255) / 256, 256, 0, stream>>>(agg, nfeat);
    scatter_add_kernel<8><<<((E << 8) + 255) / 256, 256, 0, stream>>>(h2, esrc, edst, agg, E);
    sage_layer_kernel<256><<<mblocks, 256, 0, stream>>>(h2, agg, deginv,
                                                        Ws2, Wn2, b2, g2, be2, h3, N);

    // 5) per-graph mean readout
    readout_kernel<<<G, 256, 0, stream>>>(h3, hg, npg);
}